// MixedMSEPoweImbalanceV2_773094113349
// MI455X (gfx1250) — compile-verified
//
#include <hip/hip_runtime.h>
#include <hip/hip_bf16.h>
#include <stdint.h>

// ---------------------------------------------------------------------------
// MI455X (gfx1250) implementation of MixedMSEPowerImbalance.
//
// Phases (all on `stream`, all scratch in d_ws):
//   K0: zero the (P,Q) per-node accumulator + 19 global scalar accumulators
//   K1: edge kernel  — double-buffered TDM staging of edge_attr into LDS
//                      (s_wait_tensorcnt 1 pipelining), L2-resident gathers
//                      from x, global_atomic_add_f32 scatter of P/Q
//   K2: node kernel  — dP^2+dQ^2, per-column y stats and (x-y)^2 sums;
//                      wave reduction via V_WMMA_F32_16X16X4_F32 (A=ones),
//                      block combine in LDS, atomicAdd into 19 scalars
//   K3: finalize     — ddof=1 variances, mse, loss -> d_out[0..2]
// ---------------------------------------------------------------------------

#define DEG2RAD_F 0.017453292519943295f

typedef float    v2f __attribute__((ext_vector_type(2)));
typedef float    v8f __attribute__((ext_vector_type(8)));
typedef uint32_t v4u __attribute__((ext_vector_type(4)));
typedef int      v8i __attribute__((ext_vector_type(8)));
typedef int      v4i __attribute__((ext_vector_type(4)));

#define NACC 19   // [0]=sum(dP^2+dQ^2), [1..6]=sum y_c, [7..12]=sum y_c^2, [13..18]=sum (x-y)_c^2

// ---------------------------------------------------------------------------
// Wave(32) reduction using WMMA: with A = ones(16x4) and B = [v ; 0] (4x16),
// D[m,n] = sum_k B[k,n] = v_n + v_{n+16}.  C/D VGPR0 holds, for every lane L,
// the value s_{L%16}.  Four shfl_down steps finish the 16-way fold.
// Requires EXEC == all ones (callers are fully-active, convergent waves).
// ---------------------------------------------------------------------------
__device__ __forceinline__ float wave_reduce_wmma(float v) {
    v2f a; a[0] = 1.0f; a[1] = 1.0f;   // A = ones: A[m,k] = 1 for all k
    v2f b; b[0] = v;    b[1] = 0.0f;   // B rows 0/2 carry lane values, 1/3 zero
    v8f c = {};
    v8f d = __builtin_amdgcn_wmma_f32_16x16x4_f32(
        /*neg_a=*/false, a, /*neg_b=*/false, b,
        /*c_mod=*/(short)0, c, /*reuse_a=*/false, /*reuse_b=*/false);
    float s = d[0];                    // s = v_{lane%16} + v_{lane%16+16}
    s += __shfl_down(s, 8, 32);
    s += __shfl_down(s, 4, 32);
    s += __shfl_down(s, 2, 32);
    s += __shfl_down(s, 1, 32);
    return s;                          // valid in lane 0
}

// ---------------------------------------------------------------------------
// Build a 1-D TDM descriptor and issue TENSOR_LOAD_TO_LDS for one wave's
// 32-edge slice of edge_attr (64 f32 = 256 B).  tensor_dim0 clamps the tail
// (OOB elements read as zero, never touch memory past the buffer).
// ---------------------------------------------------------------------------
__device__ __forceinline__ void tdm_stage_edge_attr(const float* ea, long long we,
                                                    int n_edges, uint32_t lds_byte_off) {
    uint64_t  ga  = (uint64_t)(uintptr_t)(ea + 2 * we);
    long long rem = (long long)n_edges - we;
    if (rem < 0) rem = 0;
    if (rem > 32) rem = 32;
    const uint32_t tdim0 = (uint32_t)(rem * 2);       // valid f32 elements

    v4u g0;
    g0[0] = 1u;                                   // count=1 (valid user D#)
    g0[1] = lds_byte_off;                         // lds_addr (bytes, LDS base 0)
    g0[2] = (uint32_t)ga;                         // global_addr[31:0]
    g0[3] = (uint32_t)(ga >> 32) | (2u << 30);    // global_addr[56:32] | type=2

    v8i g1;
    g1[0] = (int)(2u << 16);                      // data_size = 4 bytes
    g1[1] = (int)((tdim0 & 0xFFFFu) << 16);       // tensor_dim0[15:0]
    g1[2] = (int)((tdim0 >> 16) | (1u << 16));    // tensor_dim0[31:16] | tensor_dim1=1
    g1[3] = (int)(64u << 16);                     // tile_dim0 = 64 elements
    g1[4] = 1;                                    // tile_dim1 = 1
    g1[5] = 64;                                   // tensor_dim0_stride
    g1[6] = 0;
    g1[7] = 0;

    v4i gz4 = {0, 0, 0, 0};                       // 2D tensor: groups 2/3 unused
    v8i gz8 = {0, 0, 0, 0, 0, 0, 0, 0};           // clang-23 6-arg form extra group
    __builtin_amdgcn_tensor_load_to_lds(g0, g1, gz4, gz4, gz8, 0);
}

// ---------------------------------------------------------------------------
__global__ void mmpi_zero_kernel(float* __restrict__ p, int n) {
    int i = blockIdx.x * blockDim.x + threadIdx.x;
    int stride = gridDim.x * blockDim.x;
    for (; i < n; i += stride) p[i] = 0.0f;
}

// ---------------------------------------------------------------------------
// Edge kernel.  One thread per source edge; both directed contributions are
// emitted per edge.  Each wave double-buffers its edge_attr slice through LDS
// with the Tensor Data Mover: the DMA for tile t+1 is issued before tile t is
// consumed, and s_wait_tensorcnt(1) retires only the older of the two
// in-flight DMAs (TDM ops complete in order per wave).  The x-gathers and the
// 4 f32 atomics per edge resolve in the 192 MB L2 (x = 24 MB, agg = 8 MB).
// ---------------------------------------------------------------------------
__global__ void mmpi_edge_kernel(const float* __restrict__ x,
                                 const float* __restrict__ ea,
                                 const long long* __restrict__ ei,
                                 float* __restrict__ agg,
                                 int n_edges) {
    __shared__ __align__(16) float s_ea[2][8 * 64];  // 2 bufs x 8 waves x 32 edges x 2 f32
    const int wave = threadIdx.x >> 5;
    const int lane = threadIdx.x & 31;
    const long long gstride = (long long)gridDim.x * blockDim.x;
    const long long base0   = (long long)blockIdx.x * blockDim.x;
    const long long woff    = (long long)(wave << 5);

    // Prologue: stage tile 0 into buffer 0.
    if (base0 < n_edges)
        tdm_stage_edge_attr(ea, base0 + woff, n_edges, (uint32_t)(wave * 256));

    int buf = 0;
    for (long long base = base0; base < n_edges; base += gstride, buf ^= 1) {
        const long long nbase = base + gstride;
        if (nbase < n_edges) {
            // Issue next tile's DMA into the other buffer, then wait until
            // only it remains outstanding => current tile's DMA has retired.
            tdm_stage_edge_attr(ea, nbase + woff, n_edges,
                                (uint32_t)((buf ^ 1) * 2048 + wave * 256));
            __builtin_amdgcn_s_wait_tensorcnt((short)1);
        } else {
            __builtin_amdgcn_s_wait_tensorcnt((short)0);
        }

        const long long e = base + threadIdx.x;
        if (e < n_edges) {
            // Prefetch the sequential index stream ahead (HBM-bound).
            __builtin_prefetch(ei + e + 8192, 0, 1);
            __builtin_prefetch(ei + n_edges + e + 8192, 0, 1);

            const int i = (int)ei[e];
            const int j = (int)ei[n_edges + e];

            const float g = s_ea[buf][wave * 64 + 2 * lane + 0];
            const float b = s_ea[buf][wave * 64 + 2 * lane + 1];

            // x rows are 24B: 8B-aligned pair loads, served from L2.
            const float2 xi = *(const float2*)(x + 6 * (size_t)i);
            const float2 xj = *(const float2*)(x + 6 * (size_t)j);

            const float d = (xi.y - xj.y) * DEG2RAD_F;
            float sn, cs;
            __sincosf(d, &sn, &cs);
            const float vv = xi.x * xj.x;
            const float cg = cs * g, sb = sn * b;
            const float sg = sn * g, cb = cs * b;

            // direction i->j accumulates at i; j->i (d negated) at j.
            atomicAdd(&agg[2 * (size_t)i + 0], vv * (cg + sb));
            atomicAdd(&agg[2 * (size_t)i + 1], vv * (sg - cb));
            atomicAdd(&agg[2 * (size_t)j + 0], vv * (cg - sb));
            atomicAdd(&agg[2 * (size_t)j + 1], vv * (-sg - cb));
        }
    }
}

// ---------------------------------------------------------------------------
// Node kernel: power-imbalance sum + column stats, hierarchical reduction.
// ---------------------------------------------------------------------------
__global__ void mmpi_node_kernel(const float* __restrict__ x,
                                 const float* __restrict__ y,
                                 const float* __restrict__ agg,
                                 float* __restrict__ accbuf,
                                 int n_nodes) {
    float acc[NACC];
#pragma unroll
    for (int a = 0; a < NACC; ++a) acc[a] = 0.0f;

    const int stride = gridDim.x * blockDim.x;
    for (int n = blockIdx.x * blockDim.x + threadIdx.x; n < n_nodes; n += stride) {
        const float2* xr = (const float2*)(x + 6 * (size_t)n);
        const float2* yr = (const float2*)(y + 6 * (size_t)n);
        const float2  ag = *(const float2*)(agg + 2 * (size_t)n);
        const float2 x01 = xr[0], x23 = xr[1], x45 = xr[2];
        const float2 y01 = yr[0], y23 = yr[1], y45 = yr[2];

        const float dP = ag.x + x23.x;   // agg_P + x[:,2]
        const float dQ = ag.y + x23.y;   // agg_Q + x[:,3]
        acc[0] += dP * dP + dQ * dQ;

        const float xv[6] = {x01.x, x01.y, x23.x, x23.y, x45.x, x45.y};
        const float yv[6] = {y01.x, y01.y, y23.x, y23.y, y45.x, y45.y};
#pragma unroll
        for (int c = 0; c < 6; ++c) {
            const float dd = xv[c] - yv[c];
            acc[1 + c]  += yv[c];
            acc[7 + c]  += yv[c] * yv[c];
            acc[13 + c] += dd * dd;
        }
    }

    // wave reduce (WMMA fold + shuffles), then cross-wave combine in LDS
    __shared__ float sred[NACC][8];
    const int wave = threadIdx.x >> 5;
    const int lane = threadIdx.x & 31;
#pragma unroll
    for (int a = 0; a < NACC; ++a) {
        const float r = wave_reduce_wmma(acc[a]);
        if (lane == 0) sred[a][wave] = r;
    }
    __syncthreads();
    if (threadIdx.x < NACC) {
        float s = 0.0f;
#pragma unroll
        for (int w = 0; w < 8; ++w) s += sred[threadIdx.x][w];
        atomicAdd(accbuf + threadIdx.x, s);
    }
}

// ---------------------------------------------------------------------------
__global__ void mmpi_final_kernel(const float* __restrict__ acc,
                                  float* __restrict__ out,
                                  int n_nodes) {
    if (blockIdx.x != 0 || threadIdx.x != 0) return;
    const float N = (float)n_nodes;
    const float power_imb = acc[0] / N;
    float msum = 0.0f;
#pragma unroll
    for (int c = 0; c < 6; ++c) {
        const float sy  = acc[1 + c];
        const float sy2 = acc[7 + c];
        const float var = (sy2 - sy * sy / N) / (N - 1.0f);  // ddof=1
        msum += acc[13 + c] / var;                            // sum((x-y)^2)/var
    }
    const float mse  = msum / (6.0f * N);
    const float loss = 0.5f * mse + 0.5f * 0.02f * power_imb; // ALPHA/TAU
    out[0] = power_imb;
    out[1] = mse;
    out[2] = loss;
}

// ---------------------------------------------------------------------------
extern "C" void kernel_launch(void* const* d_in, const int* in_sizes, int n_in,
                              void* d_out, int out_size, void* d_ws, size_t ws_size,
                              hipStream_t stream) {
    const float*     x  = (const float*)d_in[0];       // (N_nodes, 6) f32
    const float*     ea = (const float*)d_in[1];       // (N_edges, 2) f32
    const float*     y  = (const float*)d_in[2];       // (N_nodes, 6) f32
    const long long* ei = (const long long*)d_in[3];   // (2, N_edges) i64

    const int n_nodes = in_sizes[0] / 6;
    const int n_edges = in_sizes[1] / 2;

    float* agg    = (float*)d_ws;                       // 2*n_nodes f32 (P,Q interleaved)
    float* accbuf = agg + 2 * (size_t)n_nodes;          // NACC scalar accumulators
    float* out    = (float*)d_out;

    const int nz = 2 * n_nodes + 32;
    mmpi_zero_kernel<<<1024, 256, 0, stream>>>(agg, nz);
    mmpi_edge_kernel<<<4096, 256, 0, stream>>>(x, ea, ei, agg, n_edges);
    mmpi_node_kernel<<<2048, 256, 0, stream>>>(x, y, agg, accbuf, n_nodes);
    mmpi_final_kernel<<<1, 32, 0, stream>>>(accbuf, out, n_nodes);
}